// GraphSAGENetwork_56375740727879
// MI455X (gfx1250) — compile-verified
//
#include <hip/hip_runtime.h>

typedef __attribute__((ext_vector_type(2))) float v2f;
typedef __attribute__((ext_vector_type(8))) float v8f;

#define LN_EPS 1e-5f

// ---------------------------------------------------------------- utilities
__global__ void zero_kernel(float* __restrict__ p, int n) {
    int i = blockIdx.x * blockDim.x + threadIdx.x;
    if (i < n) p[i] = 0.0f;
}

// one wave per edge: lane-strided scatter-add of source features into sum[dst]
__global__ void scatter_kernel(const float* __restrict__ h,
                               const int* __restrict__ src,
                               const int* __restrict__ dst,
                               float* __restrict__ sum,
                               float* __restrict__ cnt,
                               int E, int d, int addCnt) {
    int wave = blockIdx.x * (blockDim.x >> 5) + (threadIdx.x >> 5);
    int lane = threadIdx.x & 31;
    if (wave >= E) return;
    int s = src[wave];
    int t = dst[wave];
    const float* hp = h + (size_t)s * d;
    float*       sp = sum + (size_t)t * d;
    for (int f = lane; f < d; f += 32)
        atomicAdd(&sp[f], hp[f]);
    if (addCnt && lane == 0)
        atomicAdd(&cnt[t], 1.0f);
}

__global__ void recip_kernel(float* __restrict__ cnt, int N) {
    int i = blockIdx.x * blockDim.x + threadIdx.x;
    if (i < N) cnt[i] = 1.0f / fmaxf(cnt[i], 1.0f);
}

// sum[i] *= inv[i >> log2d]
__global__ void scale_kernel(float* __restrict__ sum, const float* __restrict__ inv,
                             int n, int log2d) {
    int i = blockIdx.x * blockDim.x + threadIdx.x;
    if (i < n) sum[i] *= inv[i >> log2d];
}

// ---------------------------------------------------------------- WMMA GEMM
#define WMMA_F32(A, B, C) \
    __builtin_amdgcn_wmma_f32_16x16x4_f32(false, (A), false, (B), (short)0, (C), false, false)

__device__ __forceinline__ void store_tile(v8f acc, float* __restrict__ out,
                                           const float* __restrict__ bias,
                                           int O, int Nrows, int mBase, int col,
                                           int mOff, int relu) {
    float bv = bias[col];
#pragma unroll
    for (int r = 0; r < 8; ++r) {
        int row = mBase + mOff + r;            // C/D layout: VGPR r -> M=r (+8 for hi lanes)
        if (row < Nrows) {
            float v = acc[r] + bv;
            if (relu) v = fmaxf(v, 0.0f);
            out[(size_t)row * O + col] = v;
        }
    }
}

// out[Nrows,O] = act( Aagg @ Wl^T + Aself @ Wr^T + bias )
// Block = 8 waves. All waves share one 64-col group (cBase); wave w owns rows
// [blockM*128 + w*16, +16). Wl/Wr k-chunks (64x32) are staged in LDS with a
// +4-float padded row stride (bank-conflict-free ds_load_b64 fragments).
// Inner loop: register double-buffer of the 10 fragments per k-step.
__global__ void __launch_bounds__(256)
sage_gemm_kernel(const float* __restrict__ Aagg,
                 const float* __restrict__ Aself,
                 const float* __restrict__ Wl,
                 const float* __restrict__ Wr,
                 const float* __restrict__ bias,
                 float* __restrict__ out,
                 int Nrows, int K, int O, int relu) {
    const int KC  = 32;                        // k-chunk
    const int LDB = 36;                        // padded LDS row stride (floats)
    __shared__ float sB[2 * 64 * LDB];         // Wl chunk | Wr chunk  (18 KB)

    int colTiles = O >> 6;
    int blockM   = blockIdx.x / colTiles;
    int cBase    = (blockIdx.x % colTiles) << 6;
    int waveId   = threadIdx.x >> 5;
    int lane     = threadIdx.x & 31;
    int mBase    = blockM * 128 + waveId * 16;

    // per-lane A row (clamped) and K-half offset for the 16x4 f32 A fragment:
    // lanes 0-15 hold K = k..k+1, lanes 16-31 hold K = k+2..k+3
    int rowA = mBase + (lane & 15);
    if (rowA > Nrows - 1) rowA = Nrows - 1;
    int kh = (lane >> 4) << 1;

    const float* aAgg  = Aagg  + (size_t)rowA * K + kh;
    const float* aSelf = Aself + (size_t)rowA * K + kh;
    const float* sBl   = sB + (lane & 15) * LDB + kh;     // + j*16*LDB per col tile
    const float* sBr   = sBl + 64 * LDB;

    v8f acc0 = {}, acc1 = {}, acc2 = {}, acc3 = {};

    for (int k0 = 0; k0 < K; k0 += KC) {
        __syncthreads();                       // previous chunk fully consumed
        // ---- cooperative stage: Wl/Wr [64 rows x 32 k] -> LDS
#pragma unroll
        for (int m = 0; m < 2; ++m) {
            const float* W = m ? Wr : Wl;
            float* dstS = sB + m * (64 * LDB);
            for (int i = threadIdx.x; i < 512; i += 256) {   // 512 float4 / matrix
                int row = i >> 3;
                int c4  = (i & 7) << 2;
                float4 v = *(const float4*)(W + (size_t)(cBase + row) * K + k0 + c4);
                float* dp = dstS + row * LDB + c4;
                *(v2f*)(dp)     = v2f{v.x, v.y};
                *(v2f*)(dp + 2) = v2f{v.z, v.w};
            }
        }
        __syncthreads();

        if (k0 + KC < K) {                     // prefetch next A chunk into L2/L0
            __builtin_prefetch(aAgg + k0 + KC, 0, 3);
            __builtin_prefetch(aSelf + k0 + KC, 0, 3);
        }

        // ---- software-pipelined inner loop: 8 k-steps of 4
        v2f cur[10], nxt[10] = {};
        cur[0] = *(const v2f*)(aAgg + k0);
        cur[1] = *(const v2f*)(aSelf + k0);
#pragma unroll
        for (int j = 0; j < 4; ++j) {
            cur[2 + j] = *(const v2f*)(sBl + j * (16 * LDB));
            cur[6 + j] = *(const v2f*)(sBr + j * (16 * LDB));
        }
#pragma unroll
        for (int s = 0; s < KC / 4; ++s) {
            int kk = (s + 1) * 4;
            if (s < KC / 4 - 1) {              // issue next step's loads first
                nxt[0] = *(const v2f*)(aAgg + k0 + kk);
                nxt[1] = *(const v2f*)(aSelf + k0 + kk);
#pragma unroll
                for (int j = 0; j < 4; ++j) {
                    nxt[2 + j] = *(const v2f*)(sBl + j * (16 * LDB) + kk);
                    nxt[6 + j] = *(const v2f*)(sBr + j * (16 * LDB) + kk);
                }
            }
            acc0 = WMMA_F32(cur[0], cur[2], acc0);
            acc1 = WMMA_F32(cur[0], cur[3], acc1);
            acc2 = WMMA_F32(cur[0], cur[4], acc2);
            acc3 = WMMA_F32(cur[0], cur[5], acc3);
            acc0 = WMMA_F32(cur[1], cur[6], acc0);
            acc1 = WMMA_F32(cur[1], cur[7], acc1);
            acc2 = WMMA_F32(cur[1], cur[8], acc2);
            acc3 = WMMA_F32(cur[1], cur[9], acc3);
#pragma unroll
            for (int q = 0; q < 10; ++q) cur[q] = nxt[q];
        }
    }

    int ncol = lane & 15;
    int mOff = (lane >> 4) << 3;               // 0 or 8
    store_tile(acc0, out, bias, O, Nrows, mBase, cBase + 0  + ncol, mOff, relu);
    store_tile(acc1, out, bias, O, Nrows, mBase, cBase + 16 + ncol, mOff, relu);
    store_tile(acc2, out, bias, O, Nrows, mBase, cBase + 32 + ncol, mOff, relu);
    store_tile(acc3, out, bias, O, Nrows, mBase, cBase + 48 + ncol, mOff, relu);
}

// --------------------------------------------------- residual + LayerNorm
// one wave per row (d == 128 -> 4 elems per lane)
__global__ void ln_kernel(float* __restrict__ out, const float* __restrict__ x,
                          const float* __restrict__ gamma, const float* __restrict__ beta,
                          int N, int d) {
    int row  = blockIdx.x * (blockDim.x >> 5) + (threadIdx.x >> 5);
    int lane = threadIdx.x & 31;
    if (row >= N) return;
    const float* xr = x + (size_t)row * d;
    float* orow = out + (size_t)row * d;

    float v[4];
    float s = 0.0f, ss = 0.0f;
#pragma unroll
    for (int i = 0; i < 4; ++i) {
        float h = orow[lane + 32 * i] + xr[lane + 32 * i];
        v[i] = h;
        s += h;
        ss += h * h;
    }
#pragma unroll
    for (int off = 16; off > 0; off >>= 1) {
        s  += __shfl_xor(s,  off, 32);
        ss += __shfl_xor(ss, off, 32);
    }
    float mu  = s / (float)d;
    float var = ss / (float)d - mu * mu;
    float inv = rsqrtf(var + LN_EPS);
#pragma unroll
    for (int i = 0; i < 4; ++i) {
        int c = lane + 32 * i;
        orow[c] = (v[i] - mu) * inv * gamma[c] + beta[c];
    }
}

// ---------------------------------------------------------------- launcher
extern "C" void kernel_launch(void* const* d_in, const int* in_sizes, int n_in,
                              void* d_out, int out_size, void* d_ws, size_t ws_size,
                              hipStream_t stream) {
    const float* x     = (const float*)d_in[0];
    const int*   ei    = (const int*)d_in[1];
    const float* W1l   = (const float*)d_in[2];
    const float* W1r   = (const float*)d_in[3];
    const float* b1    = (const float*)d_in[4];
    const float* W2l   = (const float*)d_in[5];
    const float* W2r   = (const float*)d_in[6];
    const float* b2    = (const float*)d_in[7];
    const float* gamma = (const float*)d_in[8];
    const float* beta  = (const float*)d_in[9];

    const int DIN = 128, DHID = 256, DOUT = 128;
    int N = in_sizes[0] / DIN;
    int E = in_sizes[1] / 2;
    const int* src = ei;
    const int* dst = ei + E;

    float* cnt  = (float*)d_ws;                  // [N] degree, then 1/max(deg,1)
    float* bufA = cnt + N;                       // [N,256] sum/agg buffer
    float* bufB = bufA + (size_t)N * DHID;       // [N,256] h1
    float* outp = (float*)d_out;                 // [N,128]

    auto cdiv = [](long long a, long long b) { return (int)((a + b - 1) / b); };

    // 1. zero degree + layer-1 sum region
    int nz1 = N * (1 + DIN);
    zero_kernel<<<cdiv(nz1, 256), 256, 0, stream>>>((float*)d_ws, nz1);

    // 2. scatter-add x[src] into sum1[dst] (+ degree)
    scatter_kernel<<<cdiv(E, 8), 256, 0, stream>>>(x, src, dst, bufA, cnt, E, DIN, 1);

    // 3. cnt -> 1/max(cnt,1)
    recip_kernel<<<cdiv(N, 256), 256, 0, stream>>>(cnt, N);

    // 4. sum1 -> mean
    scale_kernel<<<cdiv(N * DIN, 256), 256, 0, stream>>>(bufA, cnt, N * DIN, 7);

    // 5. h1 = relu(agg1 @ W1l^T + x @ W1r^T + b1)
    {
        int blocks = cdiv(N, 128) * (DHID / 64);
        sage_gemm_kernel<<<blocks, 256, 0, stream>>>(bufA, x, W1l, W1r, b1,
                                                     bufB, N, DIN, DHID, 1);
    }

    // 6. zero layer-2 sum
    zero_kernel<<<cdiv(N * DHID, 256), 256, 0, stream>>>(bufA, N * DHID);

    // 7. scatter-add h1[src]
    scatter_kernel<<<cdiv(E, 8), 256, 0, stream>>>(bufB, src, dst, bufA, cnt, E, DHID, 0);

    // 8. sum2 -> mean
    scale_kernel<<<cdiv(N * DHID, 256), 256, 0, stream>>>(bufA, cnt, N * DHID, 8);

    // 9. h2 = agg2 @ W2l^T + h1 @ W2r^T + b2   (into d_out)
    {
        int blocks = cdiv(N, 128) * (DOUT / 64);
        sage_gemm_kernel<<<blocks, 256, 0, stream>>>(bufA, bufB, W2l, W2r, b2,
                                                     outp, N, DHID, DOUT, 0);
    }

    // 10. out = LayerNorm(h2 + x) * gamma + beta   (in place)
    ln_kernel<<<cdiv(N, 8), 256, 0, stream>>>(outp, x, gamma, beta, N, DOUT);
}